// NLB3D_69243462746610
// MI455X (gfx1250) — compile-verified
//
#include <hip/hip_runtime.h>

typedef __attribute__((ext_vector_type(2))) float v2f;
typedef __attribute__((ext_vector_type(4))) float v4f;
typedef __attribute__((ext_vector_type(8))) float v8f;

#define PS    7
#define PAD   3
#define NB    2            // batch
#define CIN   256
#define HC    128
#define DCC   3
#define HH    48
#define WW    48
#define HWSZ  (HH * WW)    // 2304
#define KK    (DCC * PS * PS) // 147
#define BN_EPS 1e-5f

// ---------------------------------------------------------------------------
// Channel-last projection GEMM via fp32 WMMA, 16(M) x 64(N) tile per wave:
//   Out[n, m] = sum_c Wm[m, c] * In[c * strideC + n]   (per batch slice)
// A fragment loaded once per K-step and reused by 4 independent WMMAs.
// grid = (Ntot/64, M/16, NB), block = 32 (one wave)
// ---------------------------------------------------------------------------
__global__ __launch_bounds__(32)
void proj_gemm_wmma(const float* __restrict__ Wm, const float* __restrict__ In,
                    float* __restrict__ Out, int M, int Cdim,
                    long inStrideBatch, long strideC, long outStrideBatch) {
    const int lane  = threadIdx.x;      // 0..31
    const int half  = lane >> 4;        // 0: K pair {0,1}, 1: K pair {2,3}
    const int l     = lane & 15;
    const int nBase = blockIdx.x * 64;
    const int mBase = blockIdx.y * 16;
    const float* in   = In + (long)blockIdx.z * inStrideBatch + nBase + l;
    const float* arow = Wm + (long)(mBase + l) * Cdim + half * 2;
    float* out        = Out + (long)blockIdx.z * outStrideBatch;

    v8f acc[4] = {};
    for (int k = 0; k < Cdim; k += 4) {
        // A fragment (16x4): lane l holds row M=l; half selects K pair (contig pair)
        v2f a = *(const v2f*)(arow + k);
        const float* br = in + (long)(k + half * 2) * strideC;
        v2f b0; b0.x = br[0];  b0.y = br[strideC];
        v2f b1; b1.x = br[16]; b1.y = br[strideC + 16];
        v2f b2; b2.x = br[32]; b2.y = br[strideC + 32];
        v2f b3; b3.x = br[48]; b3.y = br[strideC + 48];
        acc[0] = __builtin_amdgcn_wmma_f32_16x16x4_f32(false, a, false, b0, (short)0, acc[0], false, false);
        acc[1] = __builtin_amdgcn_wmma_f32_16x16x4_f32(false, a, false, b1, (short)0, acc[1], false, false);
        acc[2] = __builtin_amdgcn_wmma_f32_16x16x4_f32(false, a, false, b2, (short)0, acc[2], false, false);
        acc[3] = __builtin_amdgcn_wmma_f32_16x16x4_f32(false, a, false, b3, (short)0, acc[3], false, false);
    }
    // D: VGPR r -> row mBase + half*8 + r, col nBase + nt*16 + l; channel-last
    #pragma unroll
    for (int nt = 0; nt < 4; ++nt) {
        float* o = out + (long)(nBase + nt * 16 + l) * M + (mBase + half * 8);
        v8f d = acc[nt];
        *(v4f*)(o)     = v4f{d[0], d[1], d[2], d[3]};
        *(v4f*)(o + 4) = v4f{d[4], d[5], d[6], d[7]};
    }
}

// ---------------------------------------------------------------------------
// Per-pixel attention over clamped 7x7x3 neighborhood. One block per pixel.
// q, psi, g channel-last [.., HC]: each key row is 128 contiguous, 512B-aligned
// floats -> b128 loads. Key offsets computed once into LDS; g rows prefetched
// (global_prefetch_b8) during score computation to overlap with softmax.
// ---------------------------------------------------------------------------
__global__ __launch_bounds__(256)
void attn_kernel(const float* __restrict__ Q, const float* __restrict__ Psi,
                 const float* __restrict__ G, float* __restrict__ Y) {
    __shared__ __align__(16) float qs[HC];
    __shared__ float sc[256];
    __shared__ float red[256];
    __shared__ int   kboff[256];        // element offset of each key row (x HC)

    const int p  = blockIdx.x;          // 0 .. NB*HWSZ-1
    const int b  = p / HWSZ;
    const int hw = p - b * HWSZ;
    const int y0 = hw / WW;
    const int x0 = hw - y0 * WW;
    const int t  = threadIdx.x;

    if (t < HC) qs[t] = Q[(long)p * HC + t];
    __syncthreads();

    float s = -3.0e38f;
    if (t < KK) {
        const int d   = t / (PS * PS);
        const int rem = t - d * (PS * PS);
        const int i   = rem / PS;
        const int j   = rem - i * PS;
        const int yy  = min(max(y0 + i - PAD, 0), HH - 1);
        const int xx  = min(max(x0 + j - PAD, 0), WW - 1);
        const int off = ((b * DCC + d) * HWSZ + yy * WW + xx) * HC;
        kboff[t] = off;
        // prefetch this key's g row (used in stage C) while scores/softmax run
        const float* gpf = G + off;
        __builtin_prefetch(gpf,      0, 0);
        __builtin_prefetch(gpf + 32, 0, 0);
        __builtin_prefetch(gpf + 64, 0, 0);
        __builtin_prefetch(gpf + 96, 0, 0);
        const v4f* pp4 = (const v4f*)(Psi + off);
        const v4f* qs4 = (const v4f*)qs;
        float acc = 0.f;
        #pragma unroll 8
        for (int h = 0; h < HC / 4; ++h) {
            v4f qv = qs4[h];
            v4f pv = pp4[h];
            acc = fmaf(qv.x, pv.x, acc);
            acc = fmaf(qv.y, pv.y, acc);
            acc = fmaf(qv.z, pv.z, acc);
            acc = fmaf(qv.w, pv.w, acc);
        }
        s = acc;
    }
    sc[t] = s;
    red[t] = s;
    __syncthreads();
    for (int off = 128; off; off >>= 1) {
        if (t < off) red[t] = fmaxf(red[t], red[t + off]);
        __syncthreads();
    }
    const float mx = red[0];
    __syncthreads();
    const float e = (t < KK) ? __expf(sc[t] - mx) : 0.f;
    sc[t] = e;
    red[t] = e;
    __syncthreads();
    for (int off = 128; off; off >>= 1) {
        if (t < off) red[t] += red[t + off];
        __syncthreads();
    }
    const float inv = 1.f / red[0];

    if (t < HC) {
        float acc = 0.f;
        for (int k = 0; k < KK; ++k)
            acc = fmaf(sc[k], G[kboff[k] + t], acc);
        Y[(long)p * HC + t] = acc * inv;
    }
}

// ---------------------------------------------------------------------------
// Output projection (fp32 WMMA, 16x64 tile) + bias; channel-first wy[b,c,hw]
// plus per-channel sum / sumsq for training-mode BatchNorm (16-lane shfl
// reduction, one atomic per channel-row per tile).
// grid = (HWSZ/64, CIN/16, NB), block = 32
// ---------------------------------------------------------------------------
__global__ __launch_bounds__(32)
void out_gemm_wmma(const float* __restrict__ Ww, const float* __restrict__ Wb,
                   const float* __restrict__ Y, float* __restrict__ Wy,
                   float* __restrict__ Sum, float* __restrict__ Sumsq) {
    const int lane  = threadIdx.x;
    const int half  = lane >> 4;
    const int l     = lane & 15;
    const int nBase = blockIdx.x * 64;
    const int oBase = blockIdx.y * 16;
    const int b     = blockIdx.z;
    // B element (k=h, n): y[(n)*HC + h] -> K-contiguous, v2f loads
    const float* yb   = Y + (long)b * HWSZ * HC + (long)(nBase + l) * HC + half * 2;
    const float* arow = Ww + (long)(oBase + l) * HC + half * 2;

    v8f acc[4] = {};
    for (int k = 0; k < HC; k += 4) {
        v2f a  = *(const v2f*)(arow + k);
        v2f b0 = *(const v2f*)(yb + k);
        v2f b1 = *(const v2f*)(yb + 16 * HC + k);
        v2f b2 = *(const v2f*)(yb + 32 * HC + k);
        v2f b3 = *(const v2f*)(yb + 48 * HC + k);
        acc[0] = __builtin_amdgcn_wmma_f32_16x16x4_f32(false, a, false, b0, (short)0, acc[0], false, false);
        acc[1] = __builtin_amdgcn_wmma_f32_16x16x4_f32(false, a, false, b1, (short)0, acc[1], false, false);
        acc[2] = __builtin_amdgcn_wmma_f32_16x16x4_f32(false, a, false, b2, (short)0, acc[2], false, false);
        acc[3] = __builtin_amdgcn_wmma_f32_16x16x4_f32(false, a, false, b3, (short)0, acc[3], false, false);
    }
    float* wyp = Wy + ((long)b * CIN + oBase + half * 8) * HWSZ + nBase + l;
    #pragma unroll
    for (int r = 0; r < 8; ++r) {
        const int o = oBase + half * 8 + r;
        const float bias = Wb[o];
        float s = 0.f, sq = 0.f;
        #pragma unroll
        for (int nt = 0; nt < 4; ++nt) {
            const float v = acc[nt][r] + bias;
            wyp[(long)r * HWSZ + nt * 16] = v;
            s += v;
            sq += v * v;
        }
        // reduce the 16 columns of this half-wave for channel o
        for (int off = 8; off; off >>= 1) {
            s  += __shfl_xor(s,  off, 16);
            sq += __shfl_xor(sq, off, 16);
        }
        if (l == 0) {
            atomicAdd(&Sum[o], s);
            atomicAdd(&Sumsq[o], sq);
        }
    }
}

// ---------------------------------------------------------------------------
__global__ void zero_kernel(float* __restrict__ p, int n) {
    int i = blockIdx.x * blockDim.x + threadIdx.x;
    if (i < n) p[i] = 0.f;
}

__global__ void bn_residual_kernel(const float* __restrict__ X,
                                   const float* __restrict__ Wy,
                                   const float* __restrict__ Sum,
                                   const float* __restrict__ Sumsq,
                                   const float* __restrict__ gamma,
                                   const float* __restrict__ beta,
                                   float* __restrict__ Out) {
    const long total = (long)NB * CIN * HWSZ;
    const long idx = (long)blockIdx.x * blockDim.x + threadIdx.x;
    if (idx >= total) return;
    const int c = (int)((idx / HWSZ) % CIN);
    const float n = (float)(NB * HWSZ);
    const float mean = Sum[c] / n;
    const float var  = Sumsq[c] / n - mean * mean;
    const float rs   = rsqrtf(var + BN_EPS);
    const float v    = (Wy[idx] - mean) * rs * gamma[c] + beta[c];
    Out[idx] = X[idx] + v;
}

// ---------------------------------------------------------------------------
extern "C" void kernel_launch(void* const* d_in, const int* in_sizes, int n_in,
                              void* d_out, int out_size, void* d_ws, size_t ws_size,
                              hipStream_t stream) {
    const float* x       = (const float*)d_in[0];   // [NB,CIN,1,HH,WW]
    const float* ctx     = (const float*)d_in[1];   // [NB,CIN,DCC,HH,WW]
    const float* theta_w = (const float*)d_in[2];   // [HC,CIN]
    const float* psi_w   = (const float*)d_in[3];   // [HC,CIN]
    const float* g_w     = (const float*)d_in[4];   // [HC,CIN]
    const float* W_w     = (const float*)d_in[5];   // [CIN,HC]
    const float* W_b     = (const float*)d_in[6];   // [CIN]
    const float* bn_g    = (const float*)d_in[7];   // [CIN]
    const float* bn_b    = (const float*)d_in[8];   // [CIN]
    float* out           = (float*)d_out;           // [NB,CIN,HH,WW]

    // workspace layout (floats)
    float* ws = (float*)d_ws;
    const long qN  = (long)NB * HWSZ * HC;          // 589,824
    const long pgN = (long)NB * DCC * HWSZ * HC;    // 1,769,472
    const long wyN = (long)NB * CIN * HWSZ;         // 1,179,648
    float* qbuf   = ws;                 // channel-last q   [b*HWSZ+hw, h]
    float* psibuf = qbuf + qN;          // channel-last psi [(b*DCC+d)*HWSZ+hw, h]
    float* gbuf   = psibuf + pgN;       // channel-last g
    float* ybuf   = gbuf + pgN;         // channel-last y
    float* wybuf  = ybuf + qN;          // channel-first wy [b, c, hw]
    float* sumb   = wybuf + wyN;        // [CIN]
    float* sumsqb = sumb + CIN;         // [CIN]

    // 1) zero BN accumulators
    zero_kernel<<<dim3(2), dim3(256), 0, stream>>>(sumb, 2 * CIN);

    // 2) q = theta_w @ x       (N = HWSZ per batch, strideC = HWSZ)
    proj_gemm_wmma<<<dim3(HWSZ / 64, HC / 16, NB), dim3(32), 0, stream>>>(
        theta_w, x, qbuf, HC, CIN,
        (long)CIN * HWSZ, (long)HWSZ, (long)HWSZ * HC);

    // 3) psi = psi_w @ ctx     (N = DCC*HWSZ per batch, strideC = DCC*HWSZ)
    proj_gemm_wmma<<<dim3(DCC * HWSZ / 64, HC / 16, NB), dim3(32), 0, stream>>>(
        psi_w, ctx, psibuf, HC, CIN,
        (long)CIN * DCC * HWSZ, (long)DCC * HWSZ, (long)DCC * HWSZ * HC);

    // 4) g = g_w @ ctx
    proj_gemm_wmma<<<dim3(DCC * HWSZ / 64, HC / 16, NB), dim3(32), 0, stream>>>(
        g_w, ctx, gbuf, HC, CIN,
        (long)CIN * DCC * HWSZ, (long)DCC * HWSZ, (long)DCC * HWSZ * HC);

    // 5) per-pixel softmax attention
    attn_kernel<<<dim3(NB * HWSZ), dim3(256), 0, stream>>>(qbuf, psibuf, gbuf, ybuf);

    // 6) wy = W_w @ y + W_b  (+ BN partial sums)
    out_gemm_wmma<<<dim3(HWSZ / 64, CIN / 16, NB), dim3(32), 0, stream>>>(
        W_w, W_b, ybuf, wybuf, sumb, sumsqb);

    // 7) BatchNorm (batch stats) + residual
    const long total = (long)NB * CIN * HWSZ;
    bn_residual_kernel<<<dim3((unsigned)((total + 255) / 256)), dim3(256), 0, stream>>>(
        x, wybuf, sumb, sumsqb, bn_g, bn_b, out);
}